// MemoryBank_2628519985195
// MI455X (gfx1250) — compile-verified
//
#include <hip/hip_runtime.h>
#include <math.h>

typedef __attribute__((ext_vector_type(2))) float v2f;
typedef __attribute__((ext_vector_type(8))) float v8f;

#define KNB    500     // neighbors per token
#define WDIM   100     // word dim
#define HDIM   400     // hidden dim
#define EPSN   1e-12f
#define NGROUP 32      // ceil(KNB/16)
#define KPAD   (NGROUP * 16)   // 512

__global__ __launch_bounds__(256)
void memorybank_attn_kernel(const float* __restrict__ word_embs,
                            const float* __restrict__ wordmem,
                            const float* __restrict__ bankmem,
                            const int*   __restrict__ idx,
                            const int*   __restrict__ lengths,
                            float*       __restrict__ out)
{
    __shared__ float s_q[WDIM];        // normalized query
    __shared__ int   s_idx[KPAD];      // neighbor rows, zero-padded to 512
    __shared__ float s_dot[KPAD];      // raw dot(q_hat, w)
    __shared__ float s_den[KPAD];      // max(||w||, eps)
    __shared__ float s_attn[KNB];      // exp(score)*mask (unnormalized)
    __shared__ float s_red[256];

    const int n    = blockIdx.x;
    const int tid  = threadIdx.x;
    const int lane = tid & 31;
    const int wave = tid >> 5;

    // ---- load neighbor indices (zero-pad tail so phase 1 is guard-free) ----
    for (int k = tid; k < KPAD; k += 256)
        s_idx[k] = (k < KNB) ? idx[(size_t)n * KNB + k] : 0;

    // ---- normalize query into LDS ----
    float e0 = (tid < WDIM) ? word_embs[(size_t)n * WDIM + tid] : 0.0f;
    s_red[tid] = e0 * e0;
    __syncthreads();
    for (int s = 128; s > 0; s >>= 1) {
        if (tid < s) s_red[tid] += s_red[tid + s];
        __syncthreads();
    }
    {
        float qinv = 1.0f / fmaxf(sqrtf(s_red[0]), EPSN);
        if (tid < WDIM) s_q[tid] = e0 * qinv;
    }
    __syncthreads();

    // ---- phase 1: cosine scores via v_wmma_f32_16x16x4_f32 ----
    // A (16x4 f32): lanes 0-15 hold M=lane, K={d,d+1}; lanes 16-31 hold M=lane-16, K={d+2,d+3}
    // B (4x16 f32): query replicated across all 16 columns -> all D columns identical
    const int half = lane >> 4;   // 0: low K pair, 1: high K pair
    const int m    = lane & 15;

    // B operands: 25 query dim-pairs, loaded from LDS once per wave
    v2f qreg[25];
    #pragma unroll
    for (int j = 0; j < 25; ++j) {
        const int off = 4 * j + half * 2;
        qreg[j].x = s_q[off];
        qreg[j].y = s_q[off + 1];
    }

    // This wave's 4 group rows, read from LDS up front (single dscnt wait)
    int rows[4];
    #pragma unroll
    for (int gi = 0; gi < 4; ++gi)
        rows[gi] = s_idx[(wave + 8 * gi) * 16 + m];

    #pragma unroll 1
    for (int gi = 0; gi < 4; ++gi) {
        const int g    = wave + 8 * gi;
        const int base = g * 16;
        const float* __restrict__ wrow = wordmem + (size_t)rows[gi] * WDIM;

        // Stage the row slice into registers. The sched_barriers force all 25
        // b64 loads to ISSUE before any consumer -> one loadcnt wait, not 25.
        v2f areg[25];
        __builtin_amdgcn_sched_barrier(0);
        #pragma unroll
        for (int j = 0; j < 25; ++j) {
            const int off = 4 * j + half * 2;
            areg[j].x = wrow[off];
            areg[j].y = wrow[off + 1];
        }
        __builtin_amdgcn_sched_barrier(0);

        v8f c = {0.f, 0.f, 0.f, 0.f, 0.f, 0.f, 0.f, 0.f};
        float ss = 0.0f;
        #pragma unroll
        for (int j = 0; j < 25; ++j) {
            c = __builtin_amdgcn_wmma_f32_16x16x4_f32(
                    /*neg_a=*/false, areg[j], /*neg_b=*/false, qreg[j],
                    /*c_mod=*/(short)0, c, /*reuse_a=*/false, /*reuse_b=*/false);
            ss = fmaf(areg[j].x, areg[j].x, ss);
            ss = fmaf(areg[j].y, areg[j].y, ss);
        }

        // lane m + lane m+16 jointly hold neighbor m's full row -> combine
        ss += __shfl_xor(ss, 16, 32);
        if (half == 0)
            s_den[base + m] = fmaxf(sqrtf(ss), EPSN);
        // D layout: VGPR r -> M=r (lanes 0-15) / M=r+8 (lanes 16-31); columns identical
        if (m == 0) {
            #pragma unroll
            for (int r = 0; r < 8; ++r)
                s_dot[base + half * 8 + r] = c[r];
        }
    }
    __syncthreads();

    // ---- phase 2: masked exp + block sum (softmax denominator) ----
    const int len = lengths[n];
    float psum = 0.0f;
    for (int k = tid; k < KNB; k += 256) {
        const float sc = s_dot[k] / s_den[k];
        const float ev = (k < len) ? expf(sc) : 0.0f;
        s_attn[k] = ev;
        psum += ev;
    }
    s_red[tid] = psum;
    __syncthreads();
    for (int s = 128; s > 0; s >>= 1) {
        if (tid < s) s_red[tid] += s_red[tid + s];
        __syncthreads();
    }
    const float inv_sum = 1.0f / s_red[0];

    // ---- phase 3: weighted gather-sum over bankmem (coalesced 1600B rows) ----
    // 10-wide batches: gather weights/rows, issue 20 independent loads, then FMA.
    float acc0 = 0.0f, acc1 = 0.0f;
    const int h0  = tid;                                   // always < 400
    const int h1  = 256 + tid;                             // valid iff tid < 144
    const int h1c = (h1 < HDIM) ? h1 : (HDIM - 1);         // clamp: no divergence in loop

    #pragma unroll 1
    for (int kb = 0; kb < KNB; kb += 10) {                 // 500 = 50 * 10
        // warm L2 ahead (speculative prefetch, safe)
        {
            const int kp = (kb + 20 < KNB) ? (kb + 20) : (KNB - 1);
            __builtin_prefetch(bankmem + (size_t)s_idx[kp] * HDIM + h0, 0, 1);
        }
        float  w[10];
        size_t ro[10];
        #pragma unroll
        for (int j = 0; j < 10; ++j) {
            w[j]  = s_attn[kb + j];
            ro[j] = (size_t)s_idx[kb + j] * HDIM;
        }
        float b0[10], b1[10];
        __builtin_amdgcn_sched_barrier(0);
        #pragma unroll
        for (int j = 0; j < 10; ++j) b0[j] = bankmem[ro[j] + h0];
        #pragma unroll
        for (int j = 0; j < 10; ++j) b1[j] = bankmem[ro[j] + h1c];
        __builtin_amdgcn_sched_barrier(0);
        #pragma unroll
        for (int j = 0; j < 10; ++j) {
            acc0 = fmaf(w[j], b0[j], acc0);
            acc1 = fmaf(w[j], b1[j], acc1);
        }
    }
    out[(size_t)n * HDIM + h0] = acc0 * inv_sum;
    if (h1 < HDIM)
        out[(size_t)n * HDIM + h1] = acc1 * inv_sum;
}

extern "C" void kernel_launch(void* const* d_in, const int* in_sizes, int n_in,
                              void* d_out, int out_size, void* d_ws, size_t ws_size,
                              hipStream_t stream)
{
    const float* word_embs = (const float*)d_in[0];
    const float* wordmem   = (const float*)d_in[1];
    const float* bankmem   = (const float*)d_in[2];
    const int*   idx       = (const int*)d_in[3];
    const int*   lengths   = (const int*)d_in[4];
    float*       out       = (float*)d_out;

    const int num_tokens = in_sizes[4];   // 2048
    dim3 grid(num_tokens), block(256);
    memorybank_attn_kernel<<<grid, block, 0, stream>>>(
        word_embs, wordmem, bankmem, idx, lengths, out);
}